// BandSplit_75247827026576
// MI455X (gfx1250) — compile-verified
//
#include <hip/hip_runtime.h>

// ---------------------------------------------------------------------------
// BandSplit: out[b,c,t,n] = sum_w xn[b,t,n,w] * fw[n,c,w] + fb[n,c]
// with xn = (x - mu(b,n)) * rsqrt(var(b,n)+eps) * nw[n,w] + nb[n,w]
// Restructured:  out = sum_w x_raw * (fw*nw*rsig)  +  bias'(b,n,c)
// ---------------------------------------------------------------------------

typedef float v2f __attribute__((ext_vector_type(2)));
typedef float v4f __attribute__((ext_vector_type(4)));
typedef float v8f __attribute__((ext_vector_type(8)));

#define NBANDS 34
#define T_TOTAL 2000
#define CH_IN 481
#define C_OUT 128
#define WPAD 120          // max band width; scale/shift rows padded to this
#define NB_B 8

__constant__ int kChOff[NBANDS] = {
    0,
    5, 9, 13, 17, 21, 25, 29, 33, 37, 41, 45, 49, 53, 57, 61, 65, 69, 73, 77,
    81, 91, 101, 111, 121, 131,
    141, 181, 221, 261, 301, 341, 381,
    421};
__constant__ int kW[NBANDS] = {
    10,
    8, 8, 8, 8, 8, 8, 8, 8, 8, 8, 8, 8, 8, 8, 8, 8, 8, 8, 8,
    20, 20, 20, 20, 20, 20,
    80, 80, 80, 80, 80, 80, 80,
    120};
__constant__ int kG[NBANDS] = {
    0,
    1, 1, 1, 1, 1, 1, 1, 1, 1, 1, 1, 1, 1, 1, 1, 1, 1, 1, 1,
    2, 2, 2, 2, 2, 2,
    3, 3, 3, 3, 3, 3, 3,
    4};
__constant__ int kJ[NBANDS] = {
    0,
    0, 1, 2, 3, 4, 5, 6, 7, 8, 9, 10, 11, 12, 13, 14, 15, 16, 17, 18,
    0, 1, 2, 3, 4, 5,
    0, 1, 2, 3, 4, 5, 6,
    0};

// ---------------------------------------------------------------------------
// Pass 1: per-(b,band) mean/var over (T, w); emit scale = rsig*nw and
//         shift = nb - mu*rsig*nw, padded to WPAD per row.
// ---------------------------------------------------------------------------
__global__ __launch_bounds__(256) void bs_stats(
    const float* __restrict__ x,
    const float* nw0, const float* nw1, const float* nw2, const float* nw3, const float* nw4,
    const float* nb0, const float* nb1, const float* nb2, const float* nb3, const float* nb4,
    float* __restrict__ scaleArr, float* __restrict__ shiftArr)
{
    const int n = blockIdx.x;   // band
    const int b = blockIdx.y;   // batch
    const int tid = threadIdx.x;
    const int w = kW[n], chOff = kChOff[n], g = kG[n], j = kJ[n];
    const int w2 = w >> 1;            // w always even
    const int H = T_TOTAL * w2;

    float s1 = 0.f, s2 = 0.f;
    for (int i = tid; i < H; i += 256) {
        const int t = i / w2;
        const int k = i - t * w2;
        const v2f v = *(const v2f*)(x + ((size_t)(b * T_TOTAL + t) * CH_IN + chOff) * 2 + 2 * k);
        s1 += v.x + v.y;
        s2 += v.x * v.x + v.y * v.y;
    }

    __shared__ float r1[256], r2[256];
    __shared__ float sh[2];
    r1[tid] = s1; r2[tid] = s2;
    __syncthreads();
    for (int off = 128; off > 0; off >>= 1) {
        if (tid < off) { r1[tid] += r1[tid + off]; r2[tid] += r2[tid + off]; }
        __syncthreads();
    }
    if (tid == 0) {
        const float cnt = (float)(T_TOTAL * w);
        const float mu = r1[0] / cnt;
        const float var = r2[0] / cnt - mu * mu;
        sh[0] = mu;
        sh[1] = rsqrtf(var + 1e-5f);
    }
    __syncthreads();

    if (tid < w) {
        const float* nwp = g == 0 ? nw0 : g == 1 ? nw1 : g == 2 ? nw2 : g == 3 ? nw3 : nw4;
        const float* nbp = g == 0 ? nb0 : g == 1 ? nb1 : g == 2 ? nb2 : g == 3 ? nb3 : nb4;
        const float nwv = nwp[j * w + tid];
        const float nbv = nbp[j * w + tid];
        const float mu = sh[0], rs = sh[1];
        const float sc = rs * nwv;
        const size_t base = ((size_t)b * NBANDS + n) * WPAD + tid;
        scaleArr[base] = sc;
        shiftArr[base] = nbv - mu * sc;
    }
}

// ---------------------------------------------------------------------------
// Pass 1b: bias'(b,n,c) = fb[n,c] + sum_w fw[n,c,w] * shift[b,n,w]
// ---------------------------------------------------------------------------
__global__ __launch_bounds__(128) void bs_bias(
    const float* fw0, const float* fw1, const float* fw2, const float* fw3, const float* fw4,
    const float* fb0, const float* fb1, const float* fb2, const float* fb3, const float* fb4,
    const float* __restrict__ shiftArr, float* __restrict__ bias2)
{
    const int n = blockIdx.x;
    const int b = blockIdx.y;
    const int c = threadIdx.x;
    const int w = kW[n], g = kG[n], j = kJ[n];
    const float* fwp = g == 0 ? fw0 : g == 1 ? fw1 : g == 2 ? fw2 : g == 3 ? fw3 : fw4;
    const float* fbp = g == 0 ? fb0 : g == 1 ? fb1 : g == 2 ? fb2 : g == 3 ? fb3 : fb4;
    const float* frow = fwp + (size_t)(j * C_OUT + c) * w;
    const float* shp = shiftArr + ((size_t)b * NBANDS + n) * WPAD;
    float s = fbp[j * C_OUT + c];
    for (int wi = 0; wi < w; ++wi) s += frow[wi] * shp[wi];
    bias2[((size_t)b * NBANDS + n) * C_OUT + c] = s;
}

// ---------------------------------------------------------------------------
// Pass 2: WMMA f32 16x16x4 GEMM per band, LDS-staged coalesced output.
// Block = 128 threads (4 wave32), covers 64 c-rows x 16 t x all 34 bands.
// ---------------------------------------------------------------------------
template <int NB, int CHOFF, int S, int W>
__device__ inline void bs_group(
    int bandBase,
    const float* __restrict__ x, const float* __restrict__ fwg,
    const float* __restrict__ scaleArr,
    float* __restrict__ stage, const float* __restrict__ biasLds,
    int b, int t0, int cBase, int cLoc0, int m16, int half, int khalf)
{
    constexpr int KT = (W + 3) / 4;
    for (int j = 0; j < NB; ++j) {
        const int band = bandBase + j;
        const int chOff = CHOFF + j * S;
        // B operand: raw x, lane's t row (t = t0 + m16), w contiguous in memory
        const float* xrow = x + ((size_t)(b * T_TOTAL + t0 + m16) * CH_IN + chOff) * 2;
        // A operand: fw row for lane's c (c = cBase + cLoc0 + m16)
        const float* arow = fwg + (size_t)(j * C_OUT + cBase + cLoc0 + m16) * W;
        const float* sc = scaleArr + ((size_t)b * NBANDS + band) * WPAD;

        v8f acc = {0.f, 0.f, 0.f, 0.f, 0.f, 0.f, 0.f, 0.f};
#pragma unroll
        for (int kt = 0; kt < KT; ++kt) {
            const int w0 = kt * 4 + khalf;      // this lane-half's K pair
            v2f a = {0.f, 0.f};
            if (w0 < W) {                        // zero-pad K tail (W=10 only)
                a = *(const v2f*)(arow + w0);    // 8B aligned (W even, w0 even)
                a.x *= sc[w0];
                a.y *= sc[w0 + 1];
            }
            const v2f bv = *(const v2f*)(xrow + w0);  // in-bounds even for K tail
            acc = __builtin_amdgcn_wmma_f32_16x16x4_f32(
                false, a, false, bv, (short)0, acc, false, false);
        }
        // D layout: VGPR r -> row m = r + half*8, col n = m16
#pragma unroll
        for (int r = 0; r < 8; ++r) {
            const int cLocal = cLoc0 + r + half * 8;
            stage[(cLocal * 16 + m16) * NBANDS + band] =
                acc[r] + biasLds[band * 64 + cLocal];
        }
    }
}

__global__ __launch_bounds__(128) void bs_gemm(
    const float* __restrict__ x,
    const float* fw0, const float* fw1, const float* fw2, const float* fw3, const float* fw4,
    const float* __restrict__ scaleArr, const float* __restrict__ bias2,
    float* __restrict__ out)
{
    const int tb = blockIdx.x;           // 0..124
    const int cBase = blockIdx.y * 64;   // 0 or 64
    const int b = blockIdx.z;            // 0..7
    const int tid = threadIdx.x;
    const int lane = tid & 31;
    const int waveId = tid >> 5;
    const int m16 = lane & 15;
    const int half = lane >> 4;
    const int khalf = half * 2;
    const int cLoc0 = waveId * 16;
    const int t0 = tb * 16;

    extern __shared__ float smem[];
    float* stage = smem;                      // 64 * 16 * 34 floats
    float* biasLds = smem + 64 * 16 * NBANDS; // 34 * 64 floats

    for (int i = tid; i < NBANDS * 64; i += 128)
        biasLds[i] = bias2[((size_t)b * NBANDS + (i >> 6)) * C_OUT + cBase + (i & 63)];
    __syncthreads();

    bs_group<1, 0, 5, 10>(0, x, fw0, scaleArr, stage, biasLds, b, t0, cBase, cLoc0, m16, half, khalf);
    bs_group<19, 5, 4, 8>(1, x, fw1, scaleArr, stage, biasLds, b, t0, cBase, cLoc0, m16, half, khalf);
    bs_group<6, 81, 10, 20>(20, x, fw2, scaleArr, stage, biasLds, b, t0, cBase, cLoc0, m16, half, khalf);
    bs_group<7, 141, 40, 80>(26, x, fw3, scaleArr, stage, biasLds, b, t0, cBase, cLoc0, m16, half, khalf);
    bs_group<1, 421, 60, 120>(33, x, fw4, scaleArr, stage, biasLds, b, t0, cBase, cLoc0, m16, half, khalf);

    __syncthreads();

    // Coalesced store: per c row, 544 contiguous floats (16 t * 34 n).
    const size_t outBase =
        ((size_t)b * C_OUT + cBase) * ((size_t)T_TOTAL * NBANDS) + (size_t)t0 * NBANDS;
    for (int i = tid; i < 64 * 136; i += 128) {
        const int c = i / 136;
        const int r = i - c * 136;
        const v4f val = *(const v4f*)(stage + c * 544 + r * 4);
        *(v4f*)(out + outBase + (size_t)c * (T_TOTAL * NBANDS) + r * 4) = val;
    }
}

// ---------------------------------------------------------------------------
extern "C" void kernel_launch(void* const* d_in, const int* in_sizes, int n_in,
                              void* d_out, int out_size, void* d_ws, size_t ws_size,
                              hipStream_t stream)
{
    (void)in_sizes; (void)n_in; (void)out_size; (void)ws_size;

    const float* x = (const float*)d_in[0];
    const float* nw[5]; const float* nb[5]; const float* fw[5]; const float* fb[5];
    for (int g = 0; g < 5; ++g) {
        nw[g] = (const float*)d_in[1 + 4 * g + 0];
        nb[g] = (const float*)d_in[1 + 4 * g + 1];
        fw[g] = (const float*)d_in[1 + 4 * g + 2];
        fb[g] = (const float*)d_in[1 + 4 * g + 3];
    }
    float* ws = (float*)d_ws;
    float* scaleArr = ws;                          // 8*34*120 floats
    float* shiftArr = ws + NB_B * NBANDS * WPAD;   // 8*34*120 floats
    float* bias2 = ws + 2 * NB_B * NBANDS * WPAD;  // 8*34*128 floats
    float* out = (float*)d_out;

    const int smemBytes = (64 * 16 * NBANDS + NBANDS * 64) * (int)sizeof(float); // 147968
    hipFuncSetAttribute((const void*)bs_gemm,
                        hipFuncAttributeMaxDynamicSharedMemorySize, smemBytes);

    bs_stats<<<dim3(NBANDS, NB_B), 256, 0, stream>>>(
        x, nw[0], nw[1], nw[2], nw[3], nw[4],
        nb[0], nb[1], nb[2], nb[3], nb[4], scaleArr, shiftArr);

    bs_bias<<<dim3(NBANDS, NB_B), 128, 0, stream>>>(
        fw[0], fw[1], fw[2], fw[3], fw[4],
        fb[0], fb[1], fb[2], fb[3], fb[4], shiftArr, bias2);

    bs_gemm<<<dim3(125, 2, NB_B), 128, smemBytes, stream>>>(
        x, fw[0], fw[1], fw[2], fw[3], fw[4], scaleArr, bias2, out);
}